// Cholesky_from_z_12781822673021
// MI455X (gfx1250) — compile-verified
//
#include <hip/hip_runtime.h>
#include <hip/hip_bf16.h>
#include <stdint.h>

#define N       128
#define NZ      (N * (N - 1) / 2)   // 8128 packed strict-lower-triangle params
#define CHUNK   32                  // columns per LDS output tile
#define LDS_PAD 33                  // row pitch of output tile (bank-conflict pad)

#if defined(__has_builtin)
#  if __has_builtin(__builtin_amdgcn_tensor_load_to_lds)
#    define HAVE_TDM 1
#  endif
#endif

typedef unsigned int u32x4 __attribute__((ext_vector_type(4)));
typedef int          i32x4 __attribute__((ext_vector_type(4)));
typedef int          i32x8 __attribute__((ext_vector_type(8)));

__global__ __launch_bounds__(N) void chol_from_z_kernel(const float* __restrict__ z,
                                                        float* __restrict__ out) {
  __shared__ float zsh[NZ];                 // staged z row: 32512 B
  __shared__ float ltile[N * LDS_PAD];      // 32-column output tile: 16896 B

  const int tid = threadIdx.x;
  const int b   = blockIdx.x;
  const float* __restrict__ zb = z + (size_t)b * NZ;
  float* __restrict__ ob       = out + (size_t)b * (size_t)(N * N);

  // ---- Stage z[b, :] into LDS via the Tensor Data Mover (gfx1250 TDM) ----
#if defined(HAVE_TDM) && defined(__gfx1250__)
  if (tid < 32) {  // one wave issues the DMA descriptor
    const uint64_t gaddr  = (uint64_t)(uintptr_t)zb;
    const uint32_t ldsoff = (uint32_t)(uintptr_t)(void*)&zsh[0];  // low 32 bits = LDS offset

    // D# group 0 (128b): count=1, lds_addr, global_addr[56:0], type=2 ("image")
    u32x4 g0;
    g0.x = 1u;                                                     // count = 1 (valid)
    g0.y = ldsoff;                                                 // lds_addr
    g0.z = (uint32_t)gaddr;                                        // global_addr[31:0]
    g0.w = (uint32_t)((gaddr >> 32) & 0x01FFFFFFu) | (2u << 30);   // global_addr[56:32], type=2

    // D# group 1 (256b): data_size=4B, 1-D tensor/tile of NZ elements
    i32x8 g1;
    g1[0] = (int)(2u << 16);                      // workgroup_mask=0, data_size=2 (4 bytes)
    g1[1] = (int)((uint32_t)NZ << 16);            // tensor_dim0[15:0] -> bits[63:48]
    g1[2] = (int)(((uint32_t)NZ >> 16) | (1u << 16)); // tensor_dim0[31:16], tensor_dim1=1
    g1[3] = (int)((uint32_t)NZ << 16);            // tensor_dim1[31:16]=0, tile_dim0=NZ
    g1[4] = 1;                                    // tile_dim1=1, tile_dim2=0
    g1[5] = NZ;                                   // tensor_dim0_stride[31:0]
    g1[6] = 0;                                    // stride[47:32]=0, tensor_dim1_stride lo=0
    g1[7] = 0;

    i32x4 z4 = {0, 0, 0, 0};
#if __clang_major__ >= 23
    i32x8 z8 = {0, 0, 0, 0, 0, 0, 0, 0};
    __builtin_amdgcn_tensor_load_to_lds(g0, g1, z4, z4, z8, 0);
#else
    __builtin_amdgcn_tensor_load_to_lds(g0, g1, z4, z4, 0);
#endif
    __builtin_amdgcn_s_wait_tensorcnt(0);         // s_wait_tensorcnt 0
  }
  __syncthreads();
#else
  for (int k = tid; k < NZ; k += N) zsh[k] = zb[k];
  __syncthreads();
#endif

  // ---- Per-row recurrence: thread i owns matrix row i ----
  const int i    = tid;
  const int base = (i * (i - 1)) >> 1;   // start of row i in packed tril layout
  float s = 0.0f;                        // running sum of squares along the row

  for (int c = 0; c < N / CHUNK; ++c) {
#pragma unroll
    for (int jj = 0; jj < CHUNK; ++jj) {
      const int j = c * CHUNK + jj;
      float v;
      if (j < i) {
        const float sq = __builtin_sqrtf(fmaxf(1.0f - s, 1e-12f));
        v = zsh[base + j] * sq;
        s = __builtin_fmaf(v, v, s);
      } else if (j == i) {
        v = __builtin_sqrtf(fmaxf(1.0f - s, 1e-12f));
      } else {
        v = 0.0f;
      }
      ltile[i * LDS_PAD + jj] = v;
    }
    __syncthreads();
    // Coalesced flush: each wave stores 32 consecutive floats of one row.
    for (int t = tid; t < N * CHUNK; t += N) {
      const int r  = t >> 5;          // row
      const int cc = t & (CHUNK - 1); // column within chunk
      ob[r * N + c * CHUNK + cc] = ltile[r * LDS_PAD + cc];
    }
    __syncthreads();
  }
}

extern "C" void kernel_launch(void* const* d_in, const int* in_sizes, int n_in,
                              void* d_out, int out_size, void* d_ws, size_t ws_size,
                              hipStream_t stream) {
  (void)n_in; (void)out_size; (void)d_ws; (void)ws_size;
  const float* z = (const float*)d_in[0];
  float* out     = (float*)d_out;
  const int B    = in_sizes[0] / NZ;   // 2048
  chol_from_z_kernel<<<dim3(B), dim3(N), 0, stream>>>(z, out);
}